// GraphCNN_82506321756401
// MI455X (gfx1250) — compile-verified
//
#include <hip/hip_runtime.h>
#include <cmath>

typedef __attribute__((ext_vector_type(2))) float v2f;
typedef __attribute__((ext_vector_type(4))) float v4f;
typedef __attribute__((ext_vector_type(8))) float v8f;

#define B_SZ 8
#define N_SZ 4096
#define D_SZ 512
#define K_SZ 512
#define O_SZ 10

// ---------------------------------------------------------------------------
// helpers
// ---------------------------------------------------------------------------
__device__ __forceinline__ float mish_f(float x) {
  float sp = (x > 20.0f) ? x : log1pf(expf(x));
  return x * tanhf(sp);
}

// CDNA5 fp32 matrix op: D(16x16) = A(16x4) * B(4x16) + C
// A lane layout: m = lane&15, K = 2*(lane>>4) + e   (e = vector element)
// B lane layout: n = lane&15, K = 2*(lane>>4) + e
// C/D layout:    n = lane&15, m = v + 8*(lane>>4)   (v = VGPR index 0..7)
__device__ __forceinline__ v8f wmma_f32x4(v2f a, v2f b, v8f c) {
  return __builtin_amdgcn_wmma_f32_16x16x4_f32(false, a, false, b, (short)0, c,
                                               false, false);
}

// ---------------------------------------------------------------------------
// small utility kernels
// ---------------------------------------------------------------------------
__global__ void k_transpose_w(const float* __restrict__ W, float* __restrict__ WT) {
  int idx = blockIdx.x * 256 + threadIdx.x;          // D*D
  int d = idx >> 9, n = idx & 511;
  WT[n * D_SZ + d] = W[idx];
}

__global__ void k_init_centers(const float* __restrict__ X, float* __restrict__ C) {
  int idx = blockIdx.x * 256 + threadIdx.x;          // B*K*D
  int b = idx / (K_SZ * D_SZ);
  int r = idx - b * (K_SZ * D_SZ);
  C[(size_t)b * K_SZ * D_SZ + r] = X[(size_t)b * N_SZ * D_SZ + r];
}

__global__ void k_c2(const float* __restrict__ C, float* __restrict__ c2) {
  int idx = blockIdx.x * 256 + threadIdx.x;          // B*K rows
  const v4f* r4 = (const v4f*)(C + (size_t)idx * D_SZ);
  float s = 0.0f;
  for (int d = 0; d < D_SZ / 4; ++d) {
    v4f t = r4[d];
    s += t.x * t.x + t.y * t.y + t.z * t.z + t.w * t.w;
  }
  c2[idx] = s;
}

__global__ void k_zero_acc(float* __restrict__ sums, int* __restrict__ counts) {
  int idx = blockIdx.x * 256 + threadIdx.x;          // B*K*D
  sums[idx] = 0.0f;
  if (idx < B_SZ * K_SZ) counts[idx] = 0;
}

__global__ void k_zero_pooled(float* __restrict__ pooled) {
  int idx = blockIdx.x * 256 + threadIdx.x;          // B*D
  pooled[idx] = 0.0f;
}

// ---------------------------------------------------------------------------
// k-means assignment: labels[i] = argmin_k ( |C_k|^2 - 2 * (X @ C^T)[i,k] )
// One wave owns a 32-row strip; column tiles processed in pairs with a 2x2
// accumulator block -> 1 global load per WMMA.
// grid = B * 16 blocks, 256 threads (8 waves -> 256 rows per block).
// ---------------------------------------------------------------------------
__global__ void __launch_bounds__(256) k_assign(const float* __restrict__ X,
                                                const float* __restrict__ C,
                                                const float* __restrict__ c2,
                                                int* __restrict__ labels) {
  int wave = threadIdx.x >> 5;
  int lane = threadIdx.x & 31;
  int ll = lane & 15, lh = lane >> 4;
  int b  = blockIdx.x >> 4;
  int rb = blockIdx.x & 15;
  int row0 = rb * 256 + wave * 32;

  const float* Xb  = X  + (size_t)b * N_SZ * D_SZ;
  const float* Cb  = C  + (size_t)b * K_SZ * D_SZ;
  const float* c2b = c2 + b * K_SZ;
  const float* arow0 = Xb + (size_t)(row0 + ll) * D_SZ + 2 * lh;
  const float* arow1 = arow0 + 16 * D_SZ;

  float best0[8], best1[8];
  int   bidx0[8], bidx1[8];
#pragma unroll
  for (int v = 0; v < 8; ++v) {
    best0[v] = 3.4e38f; bidx0[v] = 0;
    best1[v] = 3.4e38f; bidx1[v] = 0;
  }

  for (int cp = 0; cp < 16; ++cp) {                  // 16 cluster-tile pairs
    int n0 = cp * 32;
    const float* brow0 = Cb + (size_t)(n0 + ll) * D_SZ + 2 * lh;
    const float* brow1 = brow0 + 16 * D_SZ;
    v8f a00 = {0.f,0.f,0.f,0.f,0.f,0.f,0.f,0.f};
    v8f a01 = a00, a10 = a00, a11 = a00;
#pragma unroll 2
    for (int k0 = 0; k0 < D_SZ; k0 += 4) {
      v2f aA = *(const v2f*)(arow0 + k0);
      v2f aB = *(const v2f*)(arow1 + k0);
      v2f bA = *(const v2f*)(brow0 + k0);
      v2f bB = *(const v2f*)(brow1 + k0);
      a00 = wmma_f32x4(aA, bA, a00);
      a01 = wmma_f32x4(aA, bB, a01);
      a10 = wmma_f32x4(aB, bA, a10);
      a11 = wmma_f32x4(aB, bB, a11);
    }
    float c0 = c2b[n0 + ll];
    float c1 = c2b[n0 + 16 + ll];
    int   nA = n0 + ll, nB = n0 + 16 + ll;
#pragma unroll
    for (int v = 0; v < 8; ++v) {
      float d;
      d = c0 - 2.0f * a00[v]; if (d < best0[v]) { best0[v] = d; bidx0[v] = nA; }
      d = c1 - 2.0f * a01[v]; if (d < best0[v]) { best0[v] = d; bidx0[v] = nB; }
      d = c0 - 2.0f * a10[v]; if (d < best1[v]) { best1[v] = d; bidx1[v] = nA; }
      d = c1 - 2.0f * a11[v]; if (d < best1[v]) { best1[v] = d; bidx1[v] = nB; }
    }
  }
  // argmin across the 16 lanes of each half; first-index tie-breaking
#pragma unroll
  for (int v = 0; v < 8; ++v) {
    float bv = best0[v]; int bi = bidx0[v];
    for (int off = 8; off > 0; off >>= 1) {
      float ov = __shfl_xor(bv, off, 32);
      int   oi = __shfl_xor(bi, off, 32);
      if (ov < bv || (ov == bv && oi < bi)) { bv = ov; bi = oi; }
    }
    if (ll == 0) labels[b * N_SZ + row0 + v + 8 * lh] = bi;

    float cv = best1[v]; int ci = bidx1[v];
    for (int off = 8; off > 0; off >>= 1) {
      float ov = __shfl_xor(cv, off, 32);
      int   oi = __shfl_xor(ci, off, 32);
      if (ov < cv || (ov == cv && oi < ci)) { cv = ov; ci = oi; }
    }
    if (ll == 0) labels[b * N_SZ + row0 + 16 + v + 8 * lh] = ci;
  }
}

// ---------------------------------------------------------------------------
// centroid accumulation (scatter) and mean
// ---------------------------------------------------------------------------
__global__ void k_update(const float* __restrict__ X, const int* __restrict__ labels,
                         float* __restrict__ sums, int* __restrict__ counts) {
  int idx = blockIdx.x * 256 + threadIdx.x;          // B*N*D
  int d  = idx & (D_SZ - 1);
  int bi = idx >> 9;                                 // b*N + i
  int b  = bi >> 12;
  int lab = labels[bi];
  atomicAdd(&sums[((size_t)b * K_SZ + lab) * D_SZ + d], X[idx]);
  if (d == 0) atomicAdd(&counts[b * K_SZ + lab], 1);
}

__global__ void k_new_centers(const float* __restrict__ sums,
                              const int* __restrict__ counts,
                              float* __restrict__ C) {
  int idx = blockIdx.x * 256 + threadIdx.x;          // B*K*D
  int kd  = idx >> 9;                                // b*K + k
  int cnt = counts[kd];
  if (cnt > 0) C[idx] = sums[idx] / (float)cnt;      // else keep old center
}

// ---------------------------------------------------------------------------
// h = mish(adj @ onehot(labels)) as a segmented row-sum (LDS f32 atomics).
// One wave per adj row; 8 rows per block (16 KB LDS). grid = B * 512.
// ---------------------------------------------------------------------------
__global__ void __launch_bounds__(256) k_scatter_mish(const float* __restrict__ adj,
                                                      const int* __restrict__ labels,
                                                      float* __restrict__ H) {
  __shared__ float rowbuf[8][K_SZ];
  int wave = threadIdx.x >> 5, lane = threadIdx.x & 31;
  int b    = blockIdx.x >> 9;
  int rblk = blockIdx.x & 511;
  int row  = rblk * 8 + wave;

  for (int k = lane; k < K_SZ; k += 32) rowbuf[wave][k] = 0.0f;
  __syncthreads();

  const float* arow = adj + ((size_t)b * N_SZ + row) * N_SZ;
  const int*   lab  = labels + b * N_SZ;
  for (int j = lane; j < N_SZ; j += 32)
    atomicAdd(&rowbuf[wave][lab[j]], arow[j]);       // ds_add_f32
  __syncthreads();

  float* hrow = H + ((size_t)b * N_SZ + row) * D_SZ;
  for (int k = lane; k < K_SZ; k += 32) hrow[k] = mish_f(rowbuf[wave][k]);
}

// ---------------------------------------------------------------------------
// hWT[b][n][j] = (h @ W)[j][n] (stored transposed so the next GEMM's B operand
// is K-major / contiguous). 32x32 output block per wave (2x2 WMMA tiles).
// grid = B * 256 blocks of 256.
// ---------------------------------------------------------------------------
__global__ void __launch_bounds__(256) k_gemm_hw(const float* __restrict__ H,
                                                 const float* __restrict__ WT,
                                                 float* __restrict__ hWT) {
  int wave = threadIdx.x >> 5, lane = threadIdx.x & 31;
  int ll = lane & 15, lh = lane >> 4;
  int b  = blockIdx.x >> 8;
  int tb = blockIdx.x & 255;
  int wid = tb * 8 + wave;                           // 2048 blocks-of-tiles/batch
  int j0 = (wid >> 4) * 32, n0 = (wid & 15) * 32;

  const float* arow0 = H  + ((size_t)b * N_SZ + j0 + ll) * D_SZ + 2 * lh;
  const float* arow1 = arow0 + 16 * D_SZ;
  const float* brow0 = WT + (size_t)(n0 + ll) * D_SZ + 2 * lh;
  const float* brow1 = brow0 + 16 * D_SZ;

  v8f a00 = {0.f,0.f,0.f,0.f,0.f,0.f,0.f,0.f};
  v8f a01 = a00, a10 = a00, a11 = a00;
#pragma unroll 2
  for (int k0 = 0; k0 < D_SZ; k0 += 4) {
    v2f aA = *(const v2f*)(arow0 + k0);
    v2f aB = *(const v2f*)(arow1 + k0);
    v2f bA = *(const v2f*)(brow0 + k0);
    v2f bB = *(const v2f*)(brow1 + k0);
    a00 = wmma_f32x4(aA, bA, a00);
    a01 = wmma_f32x4(aA, bB, a01);
    a10 = wmma_f32x4(aB, bA, a10);
    a11 = wmma_f32x4(aB, bB, a11);
  }
  // D row m = v + 8*lh maps to hWT column j + 8*lh + v -> 8 contiguous floats
  float* o00 = hWT + ((size_t)b * D_SZ + n0 + ll) * N_SZ + j0 + 8 * lh;
  float* o01 = hWT + ((size_t)b * D_SZ + n0 + 16 + ll) * N_SZ + j0 + 8 * lh;
#pragma unroll
  for (int v = 0; v < 8; ++v) {
    o00[v] = a00[v];
    o01[v] = a01[v];
    o00[16 + v] = a10[v];
    o01[16 + v] = a11[v];
  }
}

// ---------------------------------------------------------------------------
// pooled[b][n] += sum_i mish( (adj @ hW)[i][n] + bias[n] )
// 32x32 block per wave (2x2 WMMA tiles), K = 4096; final h never materialized.
// adj streams from HBM -> prefetch it ahead; hWT strips stay cache-resident.
// grid = B * 256 blocks of 256.
// ---------------------------------------------------------------------------
__global__ void __launch_bounds__(256) k_adj_mish_pool(const float* __restrict__ adj,
                                                       const float* __restrict__ hWT,
                                                       const float* __restrict__ bias,
                                                       float* __restrict__ pooled) {
  int wave = threadIdx.x >> 5, lane = threadIdx.x & 31;
  int ll = lane & 15, lh = lane >> 4;
  int b  = blockIdx.x >> 8;
  int tb = blockIdx.x & 255;
  int wid = tb * 8 + wave;
  int i0 = (wid >> 4) * 32, n0 = (wid & 15) * 32;

  const float* arow0 = adj + ((size_t)b * N_SZ + i0 + ll) * N_SZ + 2 * lh;
  const float* arow1 = arow0 + 16 * N_SZ;
  const float* brow0 = hWT + ((size_t)b * D_SZ + n0 + ll) * N_SZ + 2 * lh;
  const float* brow1 = brow0 + 16 * N_SZ;

  v8f a00 = {0.f,0.f,0.f,0.f,0.f,0.f,0.f,0.f};
  v8f a01 = a00, a10 = a00, a11 = a00;
#pragma unroll 2
  for (int k0 = 0; k0 < N_SZ; k0 += 4) {
    if ((k0 & 63) == 0 && k0 + 256 < N_SZ) {         // warm the streaming operand
      __builtin_prefetch(arow0 + k0 + 256, 0, 1);    // global_prefetch_b8
      __builtin_prefetch(arow1 + k0 + 256, 0, 1);
    }
    v2f aA = *(const v2f*)(arow0 + k0);
    v2f aB = *(const v2f*)(arow1 + k0);
    v2f bA = *(const v2f*)(brow0 + k0);
    v2f bB = *(const v2f*)(brow1 + k0);
    a00 = wmma_f32x4(aA, bA, a00);
    a01 = wmma_f32x4(aA, bB, a01);
    a10 = wmma_f32x4(aB, bA, a10);
    a11 = wmma_f32x4(aB, bB, a11);
  }
  float bv0 = bias[n0 + ll];
  float bv1 = bias[n0 + 16 + ll];
  float p0 = 0.0f, p1 = 0.0f;
#pragma unroll
  for (int v = 0; v < 8; ++v) {
    p0 += mish_f(a00[v] + bv0) + mish_f(a10[v] + bv0);
    p1 += mish_f(a01[v] + bv1) + mish_f(a11[v] + bv1);
  }
  p0 += __shfl_xor(p0, 16, 32);                      // combine the two row halves
  p1 += __shfl_xor(p1, 16, 32);
  if (lh == 0) {
    atomicAdd(&pooled[b * D_SZ + n0 + ll], p0);
    atomicAdd(&pooled[b * D_SZ + n0 + 16 + ll], p1);
  }
}

__global__ void k_final(const float* __restrict__ pooled, const float* __restrict__ Wp,
                        const float* __restrict__ bp, float* __restrict__ out) {
  int idx = threadIdx.x;
  if (idx >= B_SZ * O_SZ) return;
  int b = idx / O_SZ, o = idx - b * O_SZ;
  float s = bp[o];
  for (int d = 0; d < D_SZ; ++d) s += pooled[b * D_SZ + d] * Wp[d * O_SZ + o];
  out[idx] = s;
}

// ---------------------------------------------------------------------------
// launch
// ---------------------------------------------------------------------------
extern "C" void kernel_launch(void* const* d_in, const int* in_sizes, int n_in,
                              void* d_out, int out_size, void* d_ws, size_t ws_size,
                              hipStream_t stream) {
  (void)in_sizes; (void)n_in; (void)out_size; (void)ws_size;
  const float* x   = (const float*)d_in[0];
  const float* adj = (const float*)d_in[1];
  const float* W   = (const float*)d_in[2];
  const float* bv  = (const float*)d_in[3];
  const float* Wp  = (const float*)d_in[4];
  const float* bp  = (const float*)d_in[5];
  float* out = (float*)d_out;

  char* w = (char*)d_ws;
  float* centers = (float*)w; w += (size_t)B_SZ * K_SZ * D_SZ * 4;   // 8 MB
  float* sums    = (float*)w; w += (size_t)B_SZ * K_SZ * D_SZ * 4;   // 8 MB
  int*   counts  = (int*)w;   w += (size_t)B_SZ * K_SZ * 4;          // 16 KB
  float* c2      = (float*)w; w += (size_t)B_SZ * K_SZ * 4;          // 16 KB
  int*   labels  = (int*)w;   w += (size_t)B_SZ * N_SZ * 4;          // 128 KB
  float* H       = (float*)w; w += (size_t)B_SZ * N_SZ * D_SZ * 4;   // 64 MB
  float* hWT     = (float*)w; w += (size_t)B_SZ * D_SZ * N_SZ * 4;   // 64 MB
  float* WT      = (float*)w; w += (size_t)D_SZ * D_SZ * 4;          // 1 MB
  float* pooled  = (float*)w; w += (size_t)B_SZ * D_SZ * 4;          // 16 KB

  k_transpose_w<<<(D_SZ * D_SZ) / 256, 256, 0, stream>>>(W, WT);

  const float* hcur = x;
  for (int layer = 0; layer < 3; ++layer) {
    k_init_centers<<<(B_SZ * K_SZ * D_SZ) / 256, 256, 0, stream>>>(hcur, centers);
    for (int it = 0; it < 10; ++it) {
      k_c2<<<(B_SZ * K_SZ) / 256, 256, 0, stream>>>(centers, c2);
      k_assign<<<B_SZ * 16, 256, 0, stream>>>(hcur, centers, c2, labels);
      k_zero_acc<<<(B_SZ * K_SZ * D_SZ) / 256, 256, 0, stream>>>(sums, counts);
      k_update<<<(B_SZ * N_SZ * D_SZ) / 256, 256, 0, stream>>>(hcur, labels, sums, counts);
      k_new_centers<<<(B_SZ * K_SZ * D_SZ) / 256, 256, 0, stream>>>(sums, counts, centers);
    }
    k_scatter_mish<<<B_SZ * 512, 256, 0, stream>>>(adj, labels, H);
    hcur = H;
  }

  k_gemm_hw<<<B_SZ * 256, 256, 0, stream>>>(H, WT, hWT);
  k_zero_pooled<<<(B_SZ * D_SZ) / 256, 256, 0, stream>>>(pooled);
  k_adj_mish_pool<<<B_SZ * 256, 256, 0, stream>>>(adj, hWT, bv, pooled);
  k_final<<<1, 128, 0, stream>>>(pooled, Wp, bp, out);
}